// SDFLookup_56307021251002
// MI455X (gfx1250) — compile-verified
//
#include <hip/hip_runtime.h>

// SDF lookup: per batch row of a (B, 120006) f32 array, gather one SDF value
// selected by integer coords computed from the 6-float tail of the row.
// Latency-bound gather; ~65 KB total traffic. No matrix FLOPs -> no WMMA/TDM.

namespace {
constexpr int kH = 200;
constexpr int kW = 200;
constexpr int kSdfSize   = kH * kW;                 // 40000
constexpr int kEndSdf    = kSdfSize;                // 40000
constexpr int kEndGrad   = kEndSdf + 2 * kSdfSize;  // 120000
constexpr int kEndRes    = kEndGrad + 2;            // 120002
constexpr int kEndOrigin = kEndRes + 2;             // 120004
constexpr int kInWidth   = kEndOrigin + 2;          // 120006
constexpr float kOobValue = -0.1f;
} // namespace

__global__ __launch_bounds__(128)
void sdf_lookup_kernel(const float* __restrict__ in,
                       float* __restrict__ out,
                       int B) {
    const int b = blockIdx.x * blockDim.x + threadIdx.x;
    if (b >= B) return;

    const float* __restrict__ row = in + (size_t)b * (size_t)kInWidth;

    // Tail of the row: res(2), origin(2), pt(2). Each pair is 8-byte aligned
    // for every b (row byte stride 480024 % 8 == 0, tail offsets % 8 == 0),
    // so three B64 loads are safe; a single B128 would be misaligned for odd b.
    const float2 res = *reinterpret_cast<const float2*>(row + kEndGrad);
    const float2 org = *reinterpret_cast<const float2*>(row + kEndRes);
    const float2 pt  = *reinterpret_cast<const float2*>(row + kEndOrigin);

    // float_offset = pt / res + origin ; truncate toward zero like astype(int32)
    const float fx = pt.x / res.x + org.x;
    const float fy = pt.y / res.y + org.y;
    const int ix = (int)fx;
    const int iy = (int)fy;

    const bool oob = (ix < 0) | (ix >= kW) | (iy < 0) | (iy >= kH);

    float v;
    if (oob) {
        v = kOobValue;   // reference overwrites the gathered value; skip the load
    } else {
        // In-bounds flat_idx is already within [0, 40000); clamp is a no-op here.
        v = row[ix * kW + iy];
    }
    out[b] = v;
}

extern "C" void kernel_launch(void* const* d_in, const int* in_sizes, int n_in,
                              void* d_out, int out_size, void* d_ws, size_t ws_size,
                              hipStream_t stream) {
    (void)n_in; (void)d_ws; (void)ws_size;
    const float* in = (const float*)d_in[0];
    float* out = (float*)d_out;

    const int B = in_sizes[0] / kInWidth;   // 2048 for the reference setup
    (void)out_size;                          // == B

    constexpr int kBlock = 128;              // 4 wave32s per block
    const int grid = (B + kBlock - 1) / kBlock;
    sdf_lookup_kernel<<<grid, kBlock, 0, stream>>>(in, out, B);
}